// RNNNet_53120155517168
// MI455X (gfx1250) — compile-verified
//
#include <hip/hip_runtime.h>
#include <math.h>

// LSTM: T=512, B=64, D=512, H=512, gates=4H=2048. fp32 throughout.
// Per-step fused kernel: gates = [h | x_t] @ [W_hh | W_ih]^T + (b_ih+b_hh),
// then pointwise LSTM cell + mask blend, using V_WMMA_F32_16X16X4_F32.
// Global->LDS staging uses CDNA5 async-to-LDS copies with double buffering
// (ASYNCcnt), so chunk ch+1 streams in while chunk ch is in the WMMA pipe.

typedef float v2f __attribute__((ext_vector_type(2)));
typedef float v8f __attribute__((ext_vector_type(8)));
typedef int   v4i __attribute__((ext_vector_type(4)));

#define T_STEPS 512
#define BATCH   64
#define DIM     512
#define HID     512

#define GLOBAL_AS __attribute__((address_space(1)))
#define LDS_AS    __attribute__((address_space(3)))

// Copy 16 bytes global -> LDS without a VGPR round-trip (ASYNCcnt-tracked).
__device__ __forceinline__ void async_copy_b128(const float* __restrict__ g,
                                                float* l) {
    __builtin_amdgcn_global_load_async_to_lds_b128(
        (GLOBAL_AS v4i*)g, (LDS_AS v4i*)l, /*offset=*/0, /*cpol=*/0);
}

template <int N>
__device__ __forceinline__ void wait_asynccnt_le() {
    __builtin_amdgcn_s_wait_asynccnt(N);   // immediate required
}

__device__ __forceinline__ float sigmoid_f(float x) {
    return 1.0f / (1.0f + __expf(-x));
}

// Grid: 16 blocks (each covers 32 hidden columns), block: 256 threads (8 waves).
// Wave w: M-tile = (w&3)*16 (covers all B=64), hidden sub-tile = (w>>2)*16.
// Each wave accumulates 4 gate tiles (i,f,g,o) of 16x16 in v8f registers.
__global__ void __launch_bounds__(256)
lstm_step_kernel(const float* __restrict__ x_t,    // [B, D]   row of inputs at t
                 const float* __restrict__ h_in,   // [B, H]
                 const float* __restrict__ c_in,   // [B, H]
                 const float* __restrict__ h0,     // [B, H] (c0 == h0)
                 const float* __restrict__ W_hh,   // [4H, H]
                 const float* __restrict__ W_ih,   // [4H, D]
                 const float* __restrict__ b_ih,   // [4H]
                 const float* __restrict__ b_hh,   // [4H]
                 const float* __restrict__ mask_t, // [B]
                 float* __restrict__ out_t,        // [B, H]
                 float* __restrict__ h_out,        // [B, H]
                 float* __restrict__ c_out)        // [B, H]
{
    // Double-buffered LDS staging. Row pitch 36 floats: 16B-aligned for B128
    // fills and conflict-free for lane-strided fragment reads.
    __shared__ float sA[2][64 * 36];          // A chunk: 64 rows x 32 K
    __shared__ float sB[2][4 * 32 * 36];      // B chunk: 4 gates x 32 N x 32 K

    const int tid  = threadIdx.x;
    const int lane = tid & 31;
    const int w    = tid >> 5;
    const int nh0  = blockIdx.x * 32;          // block's hidden-column base
    const int mt   = (w & 3) * 16;             // wave's M-tile base (batch rows)
    const int nt   = (w >> 2) * 16;            // wave's hidden sub-tile base
    const int hidx = nh0 + nt + (lane & 15);   // per-lane hidden column

    v8f acc[4];
    #pragma unroll
    for (int g = 0; g < 4; ++g) acc[g] = (v8f)0.0f;

    // Issue all async copies for chunk `ch` into buffer `buf`.
    // Each thread issues exactly 6 B128 copies (2 for A, 4 for B).
    auto issue_chunk = [&](int ch, int buf) {
        const bool   rec  = (ch < 16);
        const float* asrc = rec ? h_in : x_t;    // K 0..511 = h, 512..1023 = x
        const float* wsrc = rec ? W_hh : W_ih;
        const int    kof  = (ch & 15) * 32;
        #pragma unroll
        for (int e = tid; e < 512; e += 256) {   // sA: 64x32 = 512 float4
            const int row = e >> 3, c4 = (e & 7) * 4;
            async_copy_b128(asrc + row * 512 + kof + c4,
                            &sA[buf][row * 36 + c4]);
        }
        #pragma unroll
        for (int e = tid; e < 1024; e += 256) {  // sB: 4x32x32 = 1024 float4
            const int g = e >> 8, j = (e >> 3) & 31, c4 = (e & 7) * 4;
            async_copy_b128(wsrc + (g * 512 + nh0 + j) * 512 + kof + c4,
                            &sB[buf][(g * 32 + j) * 36 + c4]);
        }
    };

    issue_chunk(0, 0);   // prologue: stage chunk 0

    // K = 1024 in 32 chunks; chunks 0..15 use h/W_hh, 16..31 use x_t/W_ih.
    for (int ch = 0; ch < 32; ++ch) {
        const int cur = ch & 1;
        if (ch < 31) {
            issue_chunk(ch + 1, cur ^ 1);   // prefetch next chunk
            // Async loads retire in order: <=6 outstanding => chunk ch landed.
            wait_asynccnt_le<6>();
        } else {
            wait_asynccnt_le<0>();
        }
        __syncthreads();

        // f32 WMMA fragment addressing (ISA 7.12.2, 32-bit A 16x4):
        // lanes 0-15 carry K = kk,kk+1 ; lanes 16-31 carry K = kk+2,kk+3.
        const int arow = mt + (lane & 15);
        const int brow = nt + (lane & 15);
        const int ksel = (lane >> 4) * 2;

        #pragma unroll
        for (int kk = 0; kk < 32; kk += 4) {
            const v2f a = *(const v2f*)(&sA[cur][arow * 36 + kk + ksel]);
            #pragma unroll
            for (int g = 0; g < 4; ++g) {
                const v2f b = *(const v2f*)(&sB[cur][(g * 32 + brow) * 36 + kk + ksel]);
                acc[g] = __builtin_amdgcn_wmma_f32_16x16x4_f32(
                    /*neg_a=*/false, a, /*neg_b=*/false, b,
                    /*c_mod=*/(short)0, acc[g],
                    /*reuse_a=*/false, /*reuse_b=*/false);
            }
        }
        __syncthreads();   // protect buffer cur before it is refilled at ch+2
    }

    // Per-lane gate biases (all 8 C elements of a lane share one N column).
    const float bi = b_ih[0 * 512 + hidx] + b_hh[0 * 512 + hidx];
    const float bf = b_ih[1 * 512 + hidx] + b_hh[1 * 512 + hidx];
    const float bg = b_ih[2 * 512 + hidx] + b_hh[2 * 512 + hidx];
    const float bo = b_ih[3 * 512 + hidx] + b_hh[3 * 512 + hidx];
    const int mbase = mt + ((lane >> 4) << 3);   // C layout: vgpr r -> M = mbase + r

    #pragma unroll
    for (int r = 0; r < 8; ++r) {
        const int m = mbase + r;
        const float ig = sigmoid_f(acc[0][r] + bi);
        const float fg = sigmoid_f(acc[1][r] + bf);
        const float gg = tanhf(acc[2][r] + bg);
        const float og = sigmoid_f(acc[3][r] + bo);

        const float cp = c_in[m * HID + hidx];
        float cn = fg * cp + ig * gg;
        float hn = og * tanhf(cn);

        const float mk  = mask_t[m];
        const float h0v = h0[m * HID + hidx];   // c0 == h0 in the reference
        hn = hn * mk + h0v * (1.0f - mk);
        cn = cn * mk + h0v * (1.0f - mk);

        out_t[m * HID + hidx] = hn;
        h_out[m * HID + hidx] = hn;
        c_out[m * HID + hidx] = cn;
    }
}

extern "C" void kernel_launch(void* const* d_in, const int* in_sizes, int n_in,
                              void* d_out, int out_size, void* d_ws, size_t ws_size,
                              hipStream_t stream) {
    const float* inputs = (const float*)d_in[0];  // [T,B,D]
    const float* mask   = (const float*)d_in[1];  // [T,B]
    const float* h0     = (const float*)d_in[2];  // [B,H]
    const float* W_ih   = (const float*)d_in[3];  // [4H,D]
    const float* W_hh   = (const float*)d_in[4];  // [4H,H]
    const float* b_ih   = (const float*)d_in[5];  // [4H]
    const float* b_hh   = (const float*)d_in[6];  // [4H]

    float* out = (float*)d_out;                         // [T,B,H]
    float* hT  = out + (size_t)T_STEPS * BATCH * HID;   // [1,B,H]
    float* cT  = hT + (size_t)BATCH * HID;              // [1,B,H]

    // Double-buffered h/c state in workspace (4 * 64*512 floats = 512 KB).
    float* hb0 = (float*)d_ws;
    float* hb1 = hb0 + BATCH * HID;
    float* cb0 = hb1 + BATCH * HID;
    float* cb1 = cb0 + BATCH * HID;

    for (int t = 0; t < T_STEPS; ++t) {
        const bool even = ((t & 1) == 0);
        const float* h_in = (t == 0) ? h0 : (even ? hb1 : hb0);
        const float* c_in = (t == 0) ? h0 : (even ? cb1 : cb0);
        float* h_out = (t == T_STEPS - 1) ? hT : (even ? hb0 : hb1);
        float* c_out = (t == T_STEPS - 1) ? cT : (even ? cb0 : cb1);

        lstm_step_kernel<<<dim3(HID / 32), dim3(256), 0, stream>>>(
            inputs + (size_t)t * BATCH * DIM,
            h_in, c_in, h0,
            W_hh, W_ih, b_ih, b_hh,
            mask + (size_t)t * BATCH,
            out + (size_t)t * BATCH * HID,
            h_out, c_out);
    }
}